// LA_SAGE_34892314312967
// MI455X (gfx1250) — compile-verified
//
#include <hip/hip_runtime.h>
#include <cstdint>

typedef __attribute__((ext_vector_type(2))) float v2f;
typedef __attribute__((ext_vector_type(8))) float v8f;

// Problem constants (match reference)
static constexpr int cN0 = 200000, cN1 = 50000, cN2 = 10000;
static constexpr int cL = 10, cE0 = 800000, cE1 = 160000;
static constexpr int cIN = 128, cH = 256, cU = 32, cLP = 128;
static constexpr int cZK = 96; // 3*U

// ---------------------------------------------------------------- zero
__global__ void zero_kernel(float* p, size_t n) {
  size_t i = (size_t)blockIdx.x * blockDim.x + threadIdx.x;
  size_t stride = (size_t)gridDim.x * blockDim.x;
  for (; i < n; i += stride) p[i] = 0.0f;
}

// --------------------------------------------- embedding + LN + relu + concat
// one wave per node; lane handles 4 of 128 channels
__global__ __launch_bounds__(256)
void embed_ln_kernel(const int* __restrict__ x, const float* __restrict__ xu,
                     const float* __restrict__ table,
                     const float* __restrict__ g, const float* __restrict__ b,
                     float* __restrict__ h0) {
  int wave = (blockIdx.x * blockDim.x + threadIdx.x) >> 5;
  int lane = threadIdx.x & 31;
  if (wave >= cN0) return;
  const int* idx = x + (size_t)wave * cL;
  float acc[4] = {0.f, 0.f, 0.f, 0.f};
  for (int t = 0; t < cL; ++t) {
    int id = idx[t];
    if (id != 0) {  // mask zeroes embedding row 0
      const float* r = table + (size_t)id * cIN + lane * 4;
      float4 v = *(const float4*)r;
      acc[0] += v.x; acc[1] += v.y; acc[2] += v.z; acc[3] += v.w;
    }
  }
  float s = acc[0] + acc[1] + acc[2] + acc[3];
  for (int m = 16; m >= 1; m >>= 1) s += __shfl_xor(s, m, 32);
  float mean = s * (1.0f / cIN);
  float vs = 0.f;
  #pragma unroll
  for (int i = 0; i < 4; ++i) { float d = acc[i] - mean; vs += d * d; }
  for (int m = 16; m >= 1; m >>= 1) vs += __shfl_xor(vs, m, 32);
  float inv = rsqrtf(vs * (1.0f / cIN) + 1e-5f);
  float o[4];
  #pragma unroll
  for (int i = 0; i < 4; ++i) {
    int c = lane * 4 + i;
    float y = (acc[i] - mean) * inv * g[c] + b[c];
    o[i] = y > 0.f ? y : 0.f;
  }
  float* orow = h0 + (size_t)wave * (cIN + cU);
  *(float4*)(orow + lane * 4) = make_float4(o[0], o[1], o[2], o[3]);
  orow[cIN + lane] = xu[(size_t)wave * cU + lane];
}

// ------------------------------------------------------- edge gate MLP (WMMA)
// per wave: 16-edge tile. z(16x96) in LDS, w1(128x96) in LDS.
// D(16x128) = Z(16x96) @ w1^T via v_wmma_f32_16x16x4_f32, then LN+relu+w2+sigmoid.
__global__ __launch_bounds__(64)
void edge_gate_kernel(const float* __restrict__ x_src, const float* __restrict__ x_tgt,
                      const int* __restrict__ src, const int* __restrict__ dst, int E,
                      int sstride, int tstride, int un_s, int un_t,
                      const float* __restrict__ w1, const float* __restrict__ b1,
                      const float* __restrict__ g, const float* __restrict__ bn,
                      const float* __restrict__ w2, const float* __restrict__ b2p,
                      float* __restrict__ A, float* __restrict__ Asum) {
  __shared__ float w1s[cLP * cZK];     // 48 KB
  __shared__ float zs[2][16 * cZK];    // 12 KB
  int tid = threadIdx.x;
  for (int i = tid; i < cLP * cZK; i += blockDim.x) w1s[i] = w1[i];
  __syncthreads();

  int wave = tid >> 5, lane = tid & 31;
  int m = lane & 15;
  int koff = (lane >> 4) << 1;         // lanes 0-15: K pair (k,k+1); 16-31: (k+2,k+3)
  float* z = zs[wave];
  float b2 = b2p[0];

  // per-lane column constants (col = nt*16 + m)
  float c_b1[8], c_g[8], c_bn[8], c_w2[8];
  #pragma unroll
  for (int nt = 0; nt < 8; ++nt) {
    int c = nt * 16 + m;
    c_b1[nt] = b1[c]; c_g[nt] = g[c]; c_bn[nt] = bn[c]; c_w2[nt] = w2[c];
  }

  int nTiles = (E + 15) >> 4;
  for (int tile = blockIdx.x * 2 + wave; tile < nTiles; tile += gridDim.x * 2) {
    int ebase = tile * 16;
    // build z rows: lane covers one of 32 un-channels
    for (int i = 0; i < 16; ++i) {
      int e = ebase + i;
      float ui = 0.f, uj = 0.f;
      if (e < E) {
        int sj = src[e], di = dst[e];
        uj = x_src[(size_t)sj * sstride + un_s + lane];
        ui = x_tgt[(size_t)di * tstride + un_t + lane];
      }
      z[i * cZK + lane]      = fabsf(ui - uj);
      z[i * cZK + 32 + lane] = ui + uj;
      z[i * cZK + 64 + lane] = ui * uj;
    }

    v8f acc[8];
    #pragma unroll
    for (int t = 0; t < 8; ++t)
      #pragma unroll
      for (int q = 0; q < 8; ++q) acc[t][q] = 0.0f;

    const float* zrow = z + m * cZK + koff;
    for (int k = 0; k < cZK; k += 4) {
      v2f a; a.x = zrow[k]; a.y = zrow[k + 1];
      #pragma unroll
      for (int nt = 0; nt < 8; ++nt) {
        int n = nt * 16 + m;
        v2f bf; bf.x = w1s[n * cZK + k + koff]; bf.y = w1s[n * cZK + k + koff + 1];
        acc[nt] = __builtin_amdgcn_wmma_f32_16x16x4_f32(
            false, a, false, bf, (short)0, acc[nt], false, false);
      }
    }

    // rows held by this lane: r + (lane>=16 ? 8 : 0); reduce across 16-lane group
    float p[8], mean[8], inv[8];
    #pragma unroll
    for (int r = 0; r < 8; ++r) {
      float s = 0.f;
      #pragma unroll
      for (int nt = 0; nt < 8; ++nt) { float v = acc[nt][r] + c_b1[nt]; acc[nt][r] = v; s += v; }
      p[r] = s;
    }
    #pragma unroll
    for (int msk = 8; msk >= 1; msk >>= 1)
      #pragma unroll
      for (int r = 0; r < 8; ++r) p[r] += __shfl_xor(p[r], msk, 32);
    #pragma unroll
    for (int r = 0; r < 8; ++r) mean[r] = p[r] * (1.0f / cLP);
    #pragma unroll
    for (int r = 0; r < 8; ++r) {
      float s = 0.f;
      #pragma unroll
      for (int nt = 0; nt < 8; ++nt) { float d = acc[nt][r] - mean[r]; s += d * d; }
      p[r] = s;
    }
    #pragma unroll
    for (int msk = 8; msk >= 1; msk >>= 1)
      #pragma unroll
      for (int r = 0; r < 8; ++r) p[r] += __shfl_xor(p[r], msk, 32);
    #pragma unroll
    for (int r = 0; r < 8; ++r) inv[r] = rsqrtf(p[r] * (1.0f / cLP) + 1e-5f);

    // h = relu(LN*g+b); logit = h . w2
    #pragma unroll
    for (int r = 0; r < 8; ++r) {
      float s = 0.f;
      #pragma unroll
      for (int nt = 0; nt < 8; ++nt) {
        float hv = (acc[nt][r] - mean[r]) * inv[r] * c_g[nt] + c_bn[nt];
        hv = fmaxf(hv, 0.f);
        s += hv * c_w2[nt];
      }
      p[r] = s;
    }
    #pragma unroll
    for (int msk = 8; msk >= 1; msk >>= 1)
      #pragma unroll
      for (int r = 0; r < 8; ++r) p[r] += __shfl_xor(p[r], msk, 32);

    if (m == 0) {  // lane 0 -> rows 0..7, lane 16 -> rows 8..15
      int rbase = (lane >> 4) * 8;
      #pragma unroll
      for (int r = 0; r < 8; ++r) {
        int e = ebase + rbase + r;
        if (e < E) {
          float av = 1.0f / (1.0f + __expf(-(p[r] + b2)));
          A[e] = av;
          atomicAdd(&Asum[dst[e]], av);
        }
      }
    }
  }
}

// -------------------------------------------------- message scatter (atomics)
__global__ __launch_bounds__(256)
void aggregate_kernel(const float* __restrict__ xs, const int* __restrict__ src,
                      const int* __restrict__ dst, const float* __restrict__ A,
                      const float* __restrict__ Asum, float* __restrict__ agg,
                      int E, int in_c, int sstride) {
  int astride = in_c + cU;
  int wave = (blockIdx.x * blockDim.x + threadIdx.x) >> 5;
  int lane = threadIdx.x & 31;
  int nwaves = (gridDim.x * blockDim.x) >> 5;
  for (int e = wave; e < E; e += nwaves) {
    int s = src[e], d = dst[e];
    float w = A[e] / Asum[d];
    const float* xr = xs + (size_t)s * sstride;
    float* ar = agg + (size_t)d * astride;
    for (int c = lane; c < in_c; c += 32) atomicAdd(&ar[c], w * xr[c]);
    atomicAdd(&ar[in_c + lane], xr[in_c + lane]);
  }
}

// ------------------------------------------ node update GEMMs (WMMA, f32)
// out_feat = agg_feat @ llw^T + llb + tgt_feat @ lrw^T  (optional relu)
__global__ __launch_bounds__(256)
void node_update_kernel(const float* __restrict__ agg, const float* __restrict__ xt,
                        const float* __restrict__ llw, const float* __restrict__ llb,
                        const float* __restrict__ lrw, float* __restrict__ out,
                        int n_dst, int in_c, int tstride, int relu_flag) {
  int astride = in_c + cU, ostride = cH + cU;
  int wave = (blockIdx.x * blockDim.x + threadIdx.x) >> 5;
  int lane = threadIdx.x & 31;
  int nwaves = (gridDim.x * blockDim.x) >> 5;
  int m = lane & 15, koff = (lane >> 4) << 1;
  int mTiles = (n_dst + 15) >> 4;
  int total = mTiles * 4;  // 4 groups of 4 n-tiles -> 256 out channels
  for (int t = wave; t < total; t += nwaves) {
    int mt = t >> 2, ng = t & 3;
    int mbase = mt * 16, nbase = ng * 64;
    int row = mbase + m; if (row >= n_dst) row = n_dst - 1;
    const float* arow = agg + (size_t)row * astride;
    const float* trow = xt + (size_t)row * tstride;
    v8f acc[4];
    #pragma unroll
    for (int q = 0; q < 4; ++q)
      #pragma unroll
      for (int r = 0; r < 8; ++r) acc[q][r] = 0.0f;

    for (int k = 0; k < in_c; k += 4) {
      v2f a; a.x = arow[k + koff]; a.y = arow[k + koff + 1];
      #pragma unroll
      for (int q = 0; q < 4; ++q) {
        int n = nbase + q * 16 + m;
        v2f bf; bf.x = llw[(size_t)n * in_c + k + koff];
        bf.y = llw[(size_t)n * in_c + k + koff + 1];
        acc[q] = __builtin_amdgcn_wmma_f32_16x16x4_f32(
            false, a, false, bf, (short)0, acc[q], false, false);
      }
    }
    for (int k = 0; k < in_c; k += 4) {
      v2f a; a.x = trow[k + koff]; a.y = trow[k + koff + 1];
      #pragma unroll
      for (int q = 0; q < 4; ++q) {
        int n = nbase + q * 16 + m;
        v2f bf; bf.x = lrw[(size_t)n * in_c + k + koff];
        bf.y = lrw[(size_t)n * in_c + k + koff + 1];
        acc[q] = __builtin_amdgcn_wmma_f32_16x16x4_f32(
            false, a, false, bf, (short)0, acc[q], false, false);
      }
    }
    int rowoff = (lane >> 4) * 8;
    #pragma unroll
    for (int q = 0; q < 4; ++q) {
      int col = nbase + q * 16 + m;
      float bb = llb[col];
      #pragma unroll
      for (int r = 0; r < 8; ++r) {
        int orow = mbase + rowoff + r;
        if (orow < n_dst) {
          float v = acc[q][r] + bb;
          if (relu_flag) v = fmaxf(v, 0.f);
          out[(size_t)orow * ostride + col] = v;
        }
      }
    }
  }
}

// un passthrough: out[:, H:H+32] = agg[:, in_c:in_c+32]
__global__ void copy_un_kernel(const float* __restrict__ agg, float* __restrict__ out,
                               int n, int in_c) {
  int i = blockIdx.x * blockDim.x + threadIdx.x;
  int node = i >> 5, lane = i & 31;
  if (node < n)
    out[(size_t)node * (cH + cU) + cH + lane] = agg[(size_t)node * (in_c + cU) + in_c + lane];
}

// final head: out = h2[:, :256] @ out_w^T + out_b
__global__ __launch_bounds__(256)
void head_kernel(const float* __restrict__ h2, const float* __restrict__ w,
                 const float* __restrict__ bp, float* __restrict__ out, int n) {
  int wave = (blockIdx.x * blockDim.x + threadIdx.x) >> 5;
  int lane = threadIdx.x & 31;
  if (wave >= n) return;
  const float* r = h2 + (size_t)wave * (cH + cU);
  float s = 0.f;
  for (int c = lane; c < cH; c += 32) s += r[c] * w[c];
  for (int m = 16; m >= 1; m >>= 1) s += __shfl_xor(s, m, 32);
  if (lane == 0) out[wave] = s + bp[0];
}

// ------------------------------------------------------------------ launch
extern "C" void kernel_launch(void* const* d_in, const int* in_sizes, int n_in,
                              void* d_out, int out_size, void* d_ws, size_t ws_size,
                              hipStream_t stream) {
  const int*   x       = (const int*)  d_in[0];
  const float* x_unsup = (const float*)d_in[1];
  const int*   src0    = (const int*)  d_in[2];
  const int*   dst0    = (const int*)  d_in[3];
  const int*   src1    = (const int*)  d_in[4];
  const int*   dst1    = (const int*)  d_in[5];
  const float* emb     = (const float*)d_in[6];
  const float* ln_g    = (const float*)d_in[7];
  const float* ln_b    = (const float*)d_in[8];
  const float* lp_w1   = (const float*)d_in[9];
  const float* lp_b1   = (const float*)d_in[10];
  const float* lp_g    = (const float*)d_in[11];
  const float* lp_bn   = (const float*)d_in[12];
  const float* lp_w2   = (const float*)d_in[13];
  const float* lp_b2   = (const float*)d_in[14];
  const float* ll_w0   = (const float*)d_in[15];
  const float* ll_b0   = (const float*)d_in[16];
  const float* lr_w0   = (const float*)d_in[17];
  const float* ll_w1   = (const float*)d_in[18];
  const float* ll_b1   = (const float*)d_in[19];
  const float* lr_w1   = (const float*)d_in[20];
  const float* out_w   = (const float*)d_in[21];
  const float* out_b   = (const float*)d_in[22];
  float* out = (float*)d_out;

  // workspace layout (floats)
  float* ws = (float*)d_ws;
  size_t off = 0;
  float* h0    = ws + off; off += (size_t)cN0 * (cIN + cU);  // 200000 x 160
  float* agg0  = ws + off; off += (size_t)cN1 * (cIN + cU);  //  50000 x 160
  float* h1    = ws + off; off += (size_t)cN1 * (cH + cU);   //  50000 x 288
  float* agg1  = ws + off; off += (size_t)cN2 * (cH + cU);   //  10000 x 288
  float* h2    = ws + off; off += (size_t)cN2 * (cH + cU);   //  10000 x 288
  float* A0    = ws + off; off += (size_t)cE0;
  float* Asum0 = ws + off; off += (size_t)cN1;
  float* A1    = ws + off; off += (size_t)cE1;
  float* Asum1 = ws + off; off += (size_t)cN2;

  // zero accumulators (must re-zero each call: graph replays)
  zero_kernel<<<1024, 256, 0, stream>>>(agg0, (size_t)cN1 * (cIN + cU));
  zero_kernel<<<64,   256, 0, stream>>>(Asum0, (size_t)cN1);
  zero_kernel<<<512,  256, 0, stream>>>(agg1, (size_t)cN2 * (cH + cU));
  zero_kernel<<<16,   256, 0, stream>>>(Asum1, (size_t)cN2);

  // 1) embedding + LN + relu + concat -> h0 (N0 x 160)
  embed_ln_kernel<<<(cN0 * 32) / 256, 256, 0, stream>>>(x, x_unsup, emb, ln_g, ln_b, h0);

  // 2) layer 0: gate, aggregate, node update (+relu), un passthrough
  edge_gate_kernel<<<4096, 64, 0, stream>>>(h0, h0, src0, dst0, cE0,
      cIN + cU, cIN + cU, cIN, cIN,
      lp_w1, lp_b1, lp_g, lp_bn, lp_w2, lp_b2, A0, Asum0);
  aggregate_kernel<<<8192, 256, 0, stream>>>(h0, src0, dst0, A0, Asum0, agg0,
      cE0, cIN, cIN + cU);
  {
    int total = ((cN1 + 15) / 16) * 4;
    node_update_kernel<<<(total + 7) / 8, 256, 0, stream>>>(agg0, h0, ll_w0, ll_b0,
        lr_w0, h1, cN1, cIN, cIN + cU, 1 /*relu*/);
  }
  copy_un_kernel<<<(cN1 * 32) / 256, 256, 0, stream>>>(agg0, h1, cN1, cIN);

  // 3) layer 1
  edge_gate_kernel<<<2048, 64, 0, stream>>>(h1, h1, src1, dst1, cE1,
      cH + cU, cH + cU, cH, cH,
      lp_w1, lp_b1, lp_g, lp_bn, lp_w2, lp_b2, A1, Asum1);
  aggregate_kernel<<<4096, 256, 0, stream>>>(h1, src1, dst1, A1, Asum1, agg1,
      cE1, cH, cH + cU);
  {
    int total = ((cN2 + 15) / 16) * 4;
    node_update_kernel<<<(total + 7) / 8, 256, 0, stream>>>(agg1, h1, ll_w1, ll_b1,
        lr_w1, h2, cN2, cH, cH + cU, 0 /*no relu*/);
  }

  // 4) head
  head_kernel<<<(cN2 * 32) / 256, 256, 0, stream>>>(h2, out_w, out_b, out, cN2);
}